// SetAbstraction_85066122265181
// MI455X (gfx1250) — compile-verified
//
#include <hip/hip_runtime.h>
#include <hip/hip_bf16.h>
#include <math.h>

typedef float v2f __attribute__((ext_vector_type(2)));
typedef float v8f __attribute__((ext_vector_type(8)));

#define BB   16
#define NN   4096
#define SS   1024
#define KK   32
#define CINF 67
#define R2C  0.04f
#define EPSBN 1e-5f
#define MTOT (BB*SS*KK)   // 524288

// ---------------- points [B,64,N] -> pts_t [B,N,64] (tiled transpose) ---------
__global__ void transpose_pts_kernel(const float* __restrict__ points, float* __restrict__ ptst) {
  __shared__ float tile[32][33];
  int b = blockIdx.z;
  int n0 = blockIdx.x * 32;
  int c0 = blockIdx.y * 32;
  for (int j = threadIdx.y; j < 32; j += 8)
    tile[j][threadIdx.x] = points[(size_t)b*64*NN + (size_t)(c0 + j)*NN + n0 + threadIdx.x];
  __syncthreads();
  for (int j = threadIdx.y; j < 32; j += 8)
    ptst[((size_t)b*NN + n0 + j)*64 + c0 + threadIdx.x] = tile[threadIdx.x][j];
}

// ---------------- Farthest Point Sampling: 1 block per batch -----------------
__global__ void fps_kernel(const float* __restrict__ xyz, int* __restrict__ cent,
                           float* __restrict__ nxc, float* __restrict__ out_xyz) {
  int b = blockIdx.x;
  int tid = threadIdx.x;
  const float* xb = xyz + (size_t)b*3*NN;
  __shared__ int   scent[SS];
  __shared__ float swv[8];
  __shared__ int   swi[8];
  __shared__ int   sfar;

  float ld[16];                      // per-thread min-distances, i = tid + 256*j
  #pragma unroll
  for (int j = 0; j < 16; ++j) ld[j] = 1e10f;

  int far = 0;
  for (int t = 0; t < SS; ++t) {
    if (tid == 0) scent[t] = far;
    float cx = xb[far], cy = xb[NN + far], cz = xb[2*NN + far];
    float bv = -1.0f; int bi = 0;
    #pragma unroll
    for (int j = 0; j < 16; ++j) {
      int i = tid + (j << 8);
      float dx = xb[i] - cx, dy = xb[NN+i] - cy, dz = xb[2*NN+i] - cz;
      float d = dx*dx + dy*dy + dz*dz;
      float nd = fminf(ld[j], d);
      ld[j] = nd;
      if (nd > bv) { bv = nd; bi = i; }     // first-occurrence within thread
    }
    // wave32 reduce (max value, min index on tie)
    #pragma unroll
    for (int off = 16; off > 0; off >>= 1) {
      float ov = __shfl_xor(bv, off, 32);
      int   oi = __shfl_xor(bi, off, 32);
      if (ov > bv || (ov == bv && oi < bi)) { bv = ov; bi = oi; }
    }
    int wid = tid >> 5;
    if ((tid & 31) == 0) { swv[wid] = bv; swi[wid] = bi; }
    __syncthreads();
    if (tid == 0) {
      float fv = swv[0]; int fi = swi[0];
      for (int w = 1; w < 8; ++w)
        if (swv[w] > fv || (swv[w] == fv && swi[w] < fi)) { fv = swv[w]; fi = swi[w]; }
      sfar = fi;
    }
    __syncthreads();
    far = sfar;
  }
  for (int s = tid; s < SS; s += 256) {
    int p = scent[s];
    cent[b*SS + s] = p;
    float x = xb[p], y = xb[NN + p], z = xb[2*NN + p];
    size_t g = (size_t)b*SS + s;
    nxc[g*3 + 0] = x; nxc[g*3 + 1] = y; nxc[g*3 + 2] = z;
    out_xyz[(size_t)b*3*SS + 0*SS + s] = x;
    out_xyz[(size_t)b*3*SS + 1*SS + s] = y;
    out_xyz[(size_t)b*3*SS + 2*SS + s] = z;
  }
}

// ---------------- Ball query: one wave32 per centroid ------------------------
__global__ void ballquery_kernel(const float* __restrict__ xyz, const float* __restrict__ nxc,
                                 int* __restrict__ idxbuf) {
  int g = blockIdx.x * 8 + (threadIdx.x >> 5);
  int lane = threadIdx.x & 31;
  int b = g >> 10;                                  // g / SS
  const float* xb = xyz + (size_t)b*3*NN;
  float cx = nxc[(size_t)g*3+0], cy = nxc[(size_t)g*3+1], cz = nxc[(size_t)g*3+2];
  float c2 = cx*cx + cy*cy + cz*cz;
  int cnt = 0; int firstp = 0; bool havefirst = false;
  int* out = idxbuf + (size_t)g*KK;
  for (int base = 0; base < NN; base += 32) {
    int p = base + lane;
    float px = xb[p], py = xb[NN+p], pz = xb[2*NN+p];
    // same expansion formula as reference: |c|^2 + |p|^2 - 2 c.p
    float d = c2 + px*px + py*py + pz*pz - 2.0f*(cx*px + cy*py + cz*pz);
    bool inb = !(d > R2C);
    unsigned mask = (unsigned)__ballot(inb);
    if (!havefirst && mask) { firstp = base + (__ffs(mask) - 1); havefirst = true; }
    int pos = cnt + __popc(mask & ((1u << lane) - 1u));
    if (inb && pos < KK) out[pos] = p;
    cnt += __popc(mask);
    if (cnt >= KK) break;
  }
  int c = cnt < KK ? cnt : KK;
  if (lane >= c && lane < KK) out[lane] = firstp;   // pad with first in-ball index
}

// ---------------- GEMM via V_WMMA_F32_16X16X4_F32 ----------------------------
// Each wave owns a 16 x (16*NTW) output stripe: one A-fragment load feeds NTW
// WMMAs into NTW independent accumulators (dependent WMMAs spaced NTW apart).
// Weights are stored in paired-K layout so every B fragment is one contiguous,
// 8B-aligned v2f -> single ds_load_b64, no VALU repack movs. Bank-conflict-free:
//   sX pitch 68 (==4 mod 64), sW k-pair stride 2*CO+32 (==32 mod 64).
// MODE 0: A-tile gathered on the fly (layer 0 input: [dx,dy,dz, feats..., pad0])
// MODE 1: A-tile = BN(prev-layer raw output)+ReLU applied while staging to LDS
template<int KD, int CI, int CO, int MODE>
__global__ void gemm_kernel(const float* __restrict__ X, const float* __restrict__ ptst,
                            const float* __restrict__ xyz, const int* __restrict__ idxbuf,
                            const float* __restrict__ nxc,
                            const float* __restrict__ W, const float* __restrict__ bias,
                            const float* __restrict__ mu, const float* __restrict__ rstd,
                            const float* __restrict__ gam, const float* __restrict__ bet,
                            float* __restrict__ Y) {
  constexpr int NT     = CO / 16;                  // total N tiles
  constexpr int NTW    = (NT < 4) ? NT : 4;        // N tiles per wave
  constexpr int WSPLIT = NT / NTW;                 // waves sharing one M row-block
  constexpr int MWAVES = 8 / WSPLIT;               // waves along M
  constexpr int MTILE  = MWAVES * 16;              // rows per pass (128 or 64)
  constexpr int MITER  = 2;                        // passes per block (amortize sW)
  constexpr int SXP    = 68;                       // sX pitch: ==4 mod 64
  constexpr int KPS    = 2*CO + 32;                // sW k-pair stride: ==32 mod 64
  static_assert(KD <= SXP && (KD % 4) == 0, "pitch");
  __shared__ float sW[(KD/2) * KPS];
  __shared__ float sX[MTILE * SXP];
  __shared__ int   sp[MTILE];
  __shared__ float scen[MTILE * 3];

  int tid = threadIdx.x;
  for (int e = tid; e < KD * CO; e += 256) {
    int k = e / CO, n = e % CO;
    float v = (k < CI) ? W[n * CI + k] : 0.0f;           // W^T with zero K-padding
    sW[(k >> 1) * KPS + n * 2 + (k & 1)] = v;            // paired-K layout
  }

  int lane = tid & 31, wave = tid >> 5;
  int half = lane >> 4, l15 = lane & 15;
  int mw = wave / WSPLIT;                  // wave's M tile
  int n0 = (wave % WSPLIT) * NTW * 16;     // wave's first N column
  size_t mbase = (size_t)blockIdx.x * (MTILE * MITER);

  for (int it = 0; it < MITER; ++it) {
    size_t m0 = mbase + (size_t)it * MTILE;
    __syncthreads();                                   // sW ready / sX reusable
    if (MODE == 0) {
      if (tid < MTILE) {
        size_t m = m0 + tid;
        sp[tid] = idxbuf[m];
        size_t gi = m / KK;
        scen[tid*3+0] = nxc[gi*3+0];
        scen[tid*3+1] = nxc[gi*3+1];
        scen[tid*3+2] = nxc[gi*3+2];
      }
      __syncthreads();
      for (int e = tid; e < MTILE * KD; e += 256) {
        int r = e / KD, c = e % KD;
        size_t m = m0 + r;
        int p = sp[r];
        int b = (int)(m >> 15);                        // m / (SS*KK)
        float v;
        if (c < 3)        v = xyz[(size_t)b*3*NN + (size_t)c*NN + p] - scen[r*3 + c];
        else if (c < CI)  v = ptst[((size_t)b*NN + p)*64 + (c - 3)];
        else              v = 0.0f;
        sX[r * SXP + c] = v;
      }
    } else {
      for (int e = tid; e < MTILE * KD; e += 256) {
        int r = e / KD, c = e % KD;
        size_t m = m0 + r;
        float v = X[m * CI + c];
        v = (v - mu[c]) * rstd[c] * gam[c] + bet[c];
        sX[r * SXP + c] = fmaxf(v, 0.0f);
      }
    }
    __syncthreads();

    v8f acc[NTW];
    #pragma unroll
    for (int nt = 0; nt < NTW; ++nt) acc[nt] = (v8f){0.f,0.f,0.f,0.f,0.f,0.f,0.f,0.f};

    const float* arow = sX + (mw*16 + l15) * SXP + 2*half;
    const float* brow = sW + half * KPS + (n0 + l15) * 2;
    #pragma unroll
    for (int k0 = 0; k0 < KD; k0 += 4) {
      // A 16x4: lanes 0-15 K={k0,k0+1}; lanes 16-31 K={k0+2,k0+3}; contiguous v2f
      v2f a = *(const v2f*)(arow + k0);
      const float* bk = brow + (k0 >> 1) * KPS;
      #pragma unroll
      for (int nt = 0; nt < NTW; ++nt) {
        // B 4x16: same half-split along K; paired-K layout -> contiguous v2f
        v2f bfr = *(const v2f*)(bk + nt * 32);
        acc[nt] = __builtin_amdgcn_wmma_f32_16x16x4_f32(false, a, false, bfr,
                                                        (short)0, acc[nt], false, false);
      }
    }

    size_t mrow = m0 + (size_t)mw*16 + (size_t)half*8;  // D: lanes>=16 hold rows M+8
    #pragma unroll
    for (int nt = 0; nt < NTW; ++nt) {
      int col = n0 + nt*16 + l15;
      float bvv = bias[col];
      #pragma unroll
      for (int r = 0; r < 8; ++r)
        Y[(mrow + r) * CO + col] = acc[nt][r] + bvv;
    }
  }
}

// ---------------- BatchNorm stats: deterministic two-stage reduction ---------
template<int CO>
__global__ void stats_partial_kernel(const float* __restrict__ Y, float* __restrict__ part, int M) {
  constexpr int RP = 256 / CO;
  int tid = threadIdx.x;
  int c = tid % CO;
  int r = tid / CO;
  float s = 0.f, s2 = 0.f;
  for (size_t m = (size_t)blockIdx.x * RP + r; m < (size_t)M; m += (size_t)gridDim.x * RP) {
    float v = Y[m * CO + c];
    s += v; s2 += v * v;
  }
  __shared__ float ls[256], ls2[256];
  ls[tid] = s; ls2[tid] = s2;
  __syncthreads();
  for (int st = 128; st >= CO; st >>= 1) {
    if (tid < st) { ls[tid] += ls[tid + st]; ls2[tid] += ls2[tid + st]; }
    __syncthreads();
  }
  if (tid < CO) {
    part[(size_t)blockIdx.x * 2 * CO + tid]      = ls[tid];
    part[(size_t)blockIdx.x * 2 * CO + CO + tid] = ls2[tid];
  }
}

template<int CO>
__global__ void stats_final_kernel(const float* __restrict__ part, int nb, int M,
                                   float* __restrict__ mu, float* __restrict__ rstd) {
  int c = threadIdx.x;
  if (c >= CO) return;
  float s = 0.f, s2 = 0.f;
  for (int i = 0; i < nb; ++i) {
    s  += part[(size_t)i*2*CO + c];
    s2 += part[(size_t)i*2*CO + CO + c];
  }
  float m = s / (float)M;
  float var = s2 / (float)M - m * m;
  mu[c] = m;
  rstd[c] = rsqrtf(fmaxf(var, 0.f) + EPSBN);
}

// ---------------- last layer: BN + ReLU + max over K, write [B,128,S] --------
__global__ void finalmax_kernel(const float* __restrict__ Y2, const float* __restrict__ mu,
                                const float* __restrict__ rstd, const float* __restrict__ gam,
                                const float* __restrict__ bet, float* __restrict__ outp) {
  int e = blockIdx.x * 256 + threadIdx.x;   // e < B*S*128
  int c = e & 127;
  int bs = e >> 7;
  float m = mu[c], rs = rstd[c], g = gam[c], bt = bet[c];
  float mx = -1e30f;
  const float* base = Y2 + (size_t)bs * KK * 128 + c;
  #pragma unroll 4
  for (int k = 0; k < KK; ++k) {
    float v = (base[(size_t)k*128] - m) * rs * g + bt;
    v = fmaxf(v, 0.0f);
    mx = fmaxf(mx, v);
  }
  int b = bs >> 10;
  int s = bs & 1023;
  outp[(size_t)b*128*SS + (size_t)c*SS + s] = mx;
}

extern "C" void kernel_launch(void* const* d_in, const int* in_sizes, int n_in,
                              void* d_out, int out_size, void* d_ws, size_t ws_size,
                              hipStream_t stream) {
  (void)in_sizes; (void)n_in; (void)out_size; (void)ws_size;
  const float* xyz    = (const float*)d_in[0];
  const float* points = (const float*)d_in[1];
  const float* w0  = (const float*)d_in[2];
  const float* b0  = (const float*)d_in[3];
  const float* g0  = (const float*)d_in[4];
  const float* bt0 = (const float*)d_in[5];
  const float* w1  = (const float*)d_in[6];
  const float* b1  = (const float*)d_in[7];
  const float* g1  = (const float*)d_in[8];
  const float* bt1 = (const float*)d_in[9];
  const float* w2  = (const float*)d_in[10];
  const float* b2  = (const float*)d_in[11];
  const float* g2  = (const float*)d_in[12];
  const float* bt2 = (const float*)d_in[13];

  float* out      = (float*)d_out;
  float* out_xyz  = out;                        // [B,3,S]
  float* out_pts  = out + (size_t)BB*3*SS;      // [B,128,S]

  char* ws = (char*)d_ws;
  size_t off = 0;
  auto A = [&](size_t bytes) -> char* {
    char* p = ws + off;
    off += (bytes + 255) & ~(size_t)255;
    return p;
  };
  int*   cent   = (int*)  A((size_t)BB*SS*4);
  float* nxc    = (float*)A((size_t)BB*SS*3*4);
  int*   idxbuf = (int*)  A((size_t)BB*SS*KK*4);
  float* ptst   = (float*)A((size_t)BB*NN*64*4);
  float* muA    = (float*)A((size_t)3*128*4);
  float* rstdA  = (float*)A((size_t)3*128*4);
  float* part   = (float*)A((size_t)1024*2*128*4);
  float* Y0     = (float*)A((size_t)MTOT*64*4);
  float* Y1     = (float*)A((size_t)MTOT*64*4);
  float* Y2     = (float*)A((size_t)MTOT*128*4);

  transpose_pts_kernel<<<dim3(NN/32, 2, BB), dim3(32, 8), 0, stream>>>(points, ptst);
  fps_kernel<<<BB, 256, 0, stream>>>(xyz, cent, nxc, out_xyz);
  ballquery_kernel<<<BB*SS/8, 256, 0, stream>>>(xyz, nxc, idxbuf);

  // layer 0: gathered input (67 -> 64); 128 rows/pass * 2 passes -> 2048 blocks
  gemm_kernel<68, CINF, 64, 0><<<MTOT/256, 256, 0, stream>>>(
      nullptr, ptst, xyz, idxbuf, nxc, w0, b0,
      nullptr, nullptr, nullptr, nullptr, Y0);
  stats_partial_kernel<64><<<1024, 256, 0, stream>>>(Y0, part, MTOT);
  stats_final_kernel<64><<<1, 64, 0, stream>>>(part, 1024, MTOT, muA + 0, rstdA + 0);

  // layer 1: BN0+ReLU fused into load (64 -> 64)
  gemm_kernel<64, 64, 64, 1><<<MTOT/256, 256, 0, stream>>>(
      Y0, nullptr, nullptr, nullptr, nullptr, w1, b1,
      muA + 0, rstdA + 0, g0, bt0, Y1);
  stats_partial_kernel<64><<<1024, 256, 0, stream>>>(Y1, part, MTOT);
  stats_final_kernel<64><<<1, 64, 0, stream>>>(part, 1024, MTOT, muA + 128, rstdA + 128);

  // layer 2: BN1+ReLU fused into load (64 -> 128); 64 rows/pass * 2 -> 4096 blocks
  gemm_kernel<64, 64, 128, 1><<<MTOT/128, 256, 0, stream>>>(
      Y1, nullptr, nullptr, nullptr, nullptr, w2, b2,
      muA + 128, rstdA + 128, g1, bt1, Y2);
  stats_partial_kernel<128><<<1024, 256, 0, stream>>>(Y2, part, MTOT);
  stats_final_kernel<128><<<1, 128, 0, stream>>>(part, 1024, MTOT, muA + 256, rstdA + 256);

  // BN2 + ReLU + max over nsample, transpose-store
  finalmax_kernel<<<(BB*SS*128)/256, 256, 0, stream>>>(Y2, muA + 256, rstdA + 256, g2, bt2, out_pts);
}